// EncoderDecoderPreisachNNModel_18056042512593
// MI455X (gfx1250) — compile-verified
//
#include <hip/hip_runtime.h>
#include <math.h>

// ---------------- problem constants ----------------
#define NPTS   5151      // mesh points (101*102/2)
#define HID    256
#define NLAY   3
#define BATCH  8
#define TLEN   2048
#define TM     16        // mesh rows per density workgroup
#define NTILES 322       // ceil(5151/16)
#define HPAD   260       // LDS row stride (bank-conflict-free)
#define NBLK   21        // ceil(5151/256) scan blocks per batch
#define TCHUNK 64        // timesteps between barriers in scan
#define INV_TEMP 1000.0f // 1/TEMP

// workspace layout (floats)
#define WS_DENSITY 0
#define WS_DSUM    5152
#define WS_MPART   5160   // [B][NBLK][T] = 344064 floats

// output layout (floats)
#define O_BNORM   0
#define O_DENSB   16384
#define O_M       57592
#define O_S0      73976
#define O_MESHB   115184

typedef __attribute__((ext_vector_type(2))) float v2f;
typedef __attribute__((ext_vector_type(8))) float v8f;

// ---------------------------------------------------------------------------
// Kernel 1: density MLP via V_WMMA_F32_16X16X4_F32.
// One workgroup (256 thr = 8 waves) per 16-row tile of the mesh.
// ---------------------------------------------------------------------------
__global__ __launch_bounds__(256) void density_kernel(
    const float* __restrict__ mesh,   // [NPTS][2]
    const float* __restrict__ W_in,   // [2][HID]
    const float* __restrict__ b_in,   // [HID]
    const float* __restrict__ W_blk,  // [NLAY][HID][HID]
    const float* __restrict__ b_blk,  // [NLAY][HID]
    const float* __restrict__ W_out,  // [HID]
    const float* __restrict__ b_out,  // [1]
    float* __restrict__ density)      // ws [NPTS]
{
    __shared__ float hA[TM][HPAD];
    __shared__ float hB[TM][HPAD];

    const int tid  = threadIdx.x;
    const int row0 = blockIdx.x * TM;

    // ---- layer 0: h0 = relu(mesh @ W_in + b_in), scalar (K=2) ----
    for (int k = 0; k < TM; ++k) {
        const int m  = k;            // row in tile
        const int c  = tid;          // column (256 threads == 256 cols)
        const int gm = row0 + m;
        float v = 0.0f;
        if (gm < NPTS) {
            float x0 = mesh[2 * gm + 0];
            float x1 = mesh[2 * gm + 1];
            v = fmaf(x0, W_in[c], fmaf(x1, W_in[HID + c], b_in[c]));
            v = fmaxf(v, 0.0f);
        }
        hA[m][c] = v;                // padded rows -> 0
    }
    __syncthreads();

    const int wave = tid >> 5;       // 0..7
    const int lane = tid & 31;
    const int half = lane >> 4;      // 0/1
    const int l16  = lane & 15;

    float (*cur)[HPAD] = hA;
    float (*nxt)[HPAD] = hB;

    // ---- residual layers: nxt = cur + relu(cur @ W + b) ----
    for (int layer = 0; layer < NLAY; ++layer) {
        const float* __restrict__ W  = W_blk + layer * HID * HID;
        const float* __restrict__ bb = b_blk + layer * HID;

        for (int j = 0; j < 2; ++j) {
            const int n0 = (wave * 2 + j) * 16;   // output col tile
            v8f acc = {};
            #pragma unroll 4
            for (int kb = 0; kb < HID / 4; ++kb) {
                const int kbase = kb * 4 + 2 * half;
                // A fragment (16x4 f32, ISA layout: v0 K=2*half, v1 K=2*half+1)
                v2f a, b;
                a.x = cur[l16][kbase];
                a.y = cur[l16][kbase + 1];
                // B fragment (4x16 f32, symmetric layout)
                b.x = W[(kbase + 0) * HID + n0 + l16];
                b.y = W[(kbase + 1) * HID + n0 + l16];
                acc = __builtin_amdgcn_wmma_f32_16x16x4_f32(
                    false, a, false, b, (short)0, acc, false, false);
            }
            // D layout: VGPR r -> M = r + 8*half, N = n0 + l16
            const float bias = bb[n0 + l16];
            #pragma unroll
            for (int r = 0; r < 8; ++r) {
                const int m = r + 8 * half;
                float v = fmaxf(acc[r] + bias, 0.0f);
                nxt[m][n0 + l16] = cur[m][n0 + l16] + v;
            }
        }
        __syncthreads();
        float (*t)[HPAD] = cur; cur = nxt; nxt = t;
    }

    // ---- head: density = sigmoid(h @ W_out + b_out) ----
    if (tid < TM) {
        const int gm = row0 + tid;
        if (gm < NPTS) {
            float acc = b_out[0];
            #pragma unroll 8
            for (int k = 0; k < HID; ++k)
                acc = fmaf(cur[tid][k], W_out[k], acc);
            density[gm] = 1.0f / (1.0f + __expf(-acc));
        }
    }
}

// ---------------------------------------------------------------------------
// Kernel 2: sum of density (single block, deterministic)
// ---------------------------------------------------------------------------
__global__ __launch_bounds__(256) void dsum_kernel(
    const float* __restrict__ density, float* __restrict__ dsum)
{
    __shared__ float red[256];
    float s = 0.0f;
    for (int i = threadIdx.x; i < NPTS; i += 256) s += density[i];
    red[threadIdx.x] = s;
    __syncthreads();
    for (int off = 128; off > 0; off >>= 1) {
        if (threadIdx.x < off) red[threadIdx.x] += red[threadIdx.x + off];
        __syncthreads();
    }
    if (threadIdx.x == 0) dsum[0] = red[0];
}

// ---------------------------------------------------------------------------
// Kernel 3: Preisach scan. grid = (NBLK, B), 256 threads.
// Each thread owns one mesh point's state s; sequential over t.
// Per-t dot(density, s) reduced with wave32 shuffles; three barriers per 64 t.
// ---------------------------------------------------------------------------
__global__ __launch_bounds__(256) void scan_kernel(
    const float* __restrict__ decoder,  // [B][T]
    const float* __restrict__ s0,       // [B][NPTS]
    const float* __restrict__ y0,       // [B]
    const float* __restrict__ mesh,     // [NPTS][2]
    const float* __restrict__ density,  // ws [NPTS]
    float* __restrict__ m_part)         // ws [B][NBLK][T]
{
    const int b   = blockIdx.y;
    const int nb  = blockIdx.x;
    const int tid = threadIdx.x;
    const int n   = nb * 256 + tid;
    const bool valid = (n < NPTS);

    float s     = valid ? s0[b * NPTS + n] : 0.0f;
    float beta  = valid ? mesh[2 * n + 0]  : 2.0f;
    float alpha = valid ? mesh[2 * n + 1]  : 2.0f;
    float dens  = valid ? density[n]       : 0.0f;
    float h_prev = y0[b];

    const float* __restrict__ hseq = decoder + b * TLEN;
    float* __restrict__ out = m_part + (b * NBLK + nb) * TLEN;

    __shared__ float wsum[8][TCHUNK];
    __shared__ float hbuf[TCHUNK];
    const int wave = tid >> 5;
    const int lane = tid & 31;

    for (int t0 = 0; t0 < TLEN; t0 += TCHUNK) {
        if (tid < TCHUNK) {
            hbuf[tid] = hseq[t0 + tid];
            if (t0 + TCHUNK < TLEN)
                __builtin_prefetch(&hseq[t0 + TCHUNK + tid], 0, 0);
        }
        __syncthreads();

        for (int tt = 0; tt < TCHUNK; ++tt) {
            const float h  = hbuf[tt];
            // sigmoid((h-a)/TEMP) = 1/(1+exp((a-h)/TEMP))
            const float up = 1.0f / (1.0f + __expf((alpha - h) * INV_TEMP));
            const float dn = 1.0f / (1.0f + __expf((beta  - h) * INV_TEMP));
            const float s_up = fmaf(1.0f - s, up, s);          // s + (1-s)*up
            const float s_dn = fmaf(s + 1.0f, dn, -1.0f);      // -1 + (s+1)*dn
            s = (h >= h_prev) ? s_up : s_dn;
            h_prev = h;

            float c = dens * s;
            #pragma unroll
            for (int off = 16; off > 0; off >>= 1)
                c += __shfl_xor(c, off, 32);
            if (lane == 0) wsum[wave][tt] = c;
        }
        __syncthreads();

        if (tid < TCHUNK) {
            float acc = 0.0f;
            #pragma unroll
            for (int w = 0; w < 8; ++w) acc += wsum[w][tid];
            out[t0 + tid] = acc;
        }
        __syncthreads();
    }
}

// ---------------------------------------------------------------------------
// Kernel 4: finalize m and b_norm (deterministic cross-block reduce)
// ---------------------------------------------------------------------------
__global__ __launch_bounds__(256) void finalize_kernel(
    const float* __restrict__ m_part, const float* __restrict__ dsum,
    float* __restrict__ out_bnorm, float* __restrict__ out_m)
{
    const int i = blockIdx.x * 256 + threadIdx.x;   // over B*T
    if (i >= BATCH * TLEN) return;
    const int b = i / TLEN, t = i % TLEN;
    float acc = 0.0f;
    #pragma unroll
    for (int k = 0; k < NBLK; ++k)
        acc += m_part[(b * NBLK + k) * TLEN + t];
    const float m = acc / dsum[0];
    out_m[i]     = m;
    out_bnorm[i] = fmaf(0.5f, m, 0.5f);   // H_SCALE=0, M_SCALE=0.5, M_OFFSET=0.5
}

// ---------------------------------------------------------------------------
// Kernel 5: broadcast copies (density_b, initial_states, mesh_b)
// ---------------------------------------------------------------------------
__global__ __launch_bounds__(256) void copy_kernel(
    const float* __restrict__ density, const float* __restrict__ s0,
    const float* __restrict__ mesh,
    float* __restrict__ out_densb, float* __restrict__ out_s0,
    float* __restrict__ out_meshb)
{
    const int i = blockIdx.x * 256 + threadIdx.x;
    if (i < BATCH * NPTS) {
        out_densb[i] = density[i % NPTS];
        out_s0[i]    = s0[i];
    }
    if (i < BATCH * NPTS * 2) {
        out_meshb[i] = mesh[i % (NPTS * 2)];
    }
}

// ---------------------------------------------------------------------------
extern "C" void kernel_launch(void* const* d_in, const int* in_sizes, int n_in,
                              void* d_out, int out_size, void* d_ws, size_t ws_size,
                              hipStream_t stream)
{
    (void)in_sizes; (void)n_in; (void)out_size; (void)ws_size;
    // inputs per setup_inputs() order
    const float* dec   = (const float*)d_in[1];   // [8][2048][1]
    const float* s0    = (const float*)d_in[2];   // [8][5151]
    const float* y0    = (const float*)d_in[3];   // [8]
    const float* mesh  = (const float*)d_in[4];   // [5151][2]
    const float* W_in  = (const float*)d_in[5];
    const float* b_in  = (const float*)d_in[6];
    const float* W_blk = (const float*)d_in[7];
    const float* b_blk = (const float*)d_in[8];
    const float* W_out = (const float*)d_in[9];
    const float* b_out = (const float*)d_in[10];

    float* ws       = (float*)d_ws;
    float* density  = ws + WS_DENSITY;
    float* dsum     = ws + WS_DSUM;
    float* m_part   = ws + WS_MPART;

    float* out      = (float*)d_out;
    float* o_bnorm  = out + O_BNORM;
    float* o_densb  = out + O_DENSB;
    float* o_m      = out + O_M;
    float* o_s0     = out + O_S0;
    float* o_meshb  = out + O_MESHB;

    density_kernel<<<NTILES, 256, 0, stream>>>(
        mesh, W_in, b_in, W_blk, b_blk, W_out, b_out, density);
    dsum_kernel<<<1, 256, 0, stream>>>(density, dsum);
    scan_kernel<<<dim3(NBLK, BATCH), 256, 0, stream>>>(
        dec, s0, y0, mesh, density, m_part);
    finalize_kernel<<<(BATCH * TLEN + 255) / 256, 256, 0, stream>>>(
        m_part, dsum, o_bnorm, o_m);
    copy_kernel<<<(BATCH * NPTS * 2 + 255) / 256, 256, 0, stream>>>(
        density, s0, mesh, o_densb, o_s0, o_meshb);
}